// SupplyChainGNN_13821204758583
// MI455X (gfx1250) — compile-verified
//
#include <hip/hip_runtime.h>
#include <math.h>

typedef __attribute__((ext_vector_type(2))) float v2f;
typedef __attribute__((ext_vector_type(8))) float v8f;

#define HEADS 4
#define CH 64
#define HCW 256          // heads * channels
#define LRELU_SLOPE 0.2f
#define BN_EPS 1e-5f

// ---------------------------------------------------------------------------
// GEMM: Hout[N, 256] = X[N,K] @ W[K,256] using V_WMMA_F32_16X16X4_F32.
// One wave (32 lanes) per 16x16 output tile.
// A (16x4, MxK): lane&15 = M; lane half selects K-pair {0,1} vs {2,3}.
// B (4x16, KxN): lane&15 = N; lane half selects K-pair.
// C/D (16x16):   VGPR r, lanes 0-15 -> M=r, lanes 16-31 -> M=r+8; N=lane&15.
// ---------------------------------------------------------------------------
__global__ __launch_bounds__(32)
void gemm_wmma_f32(const float* __restrict__ X, const float* __restrict__ W,
                   float* __restrict__ Hout, int N, int K) {
    const int lane = threadIdx.x;
    const int half = lane >> 4;           // 0 or 1
    const int l16  = lane & 15;
    const int m0 = blockIdx.x * 16;
    const int n0 = blockIdx.y * 16;
    int m = m0 + l16;
    int mc = m < N ? m : (N - 1);         // clamp reads (EXEC must stay all-1s for WMMA)
    const int n = n0 + l16;
    const float* xrow = X + (size_t)mc * K;

    v8f acc = {0.f, 0.f, 0.f, 0.f, 0.f, 0.f, 0.f, 0.f};
    for (int k = 0; k < K; k += 4) {
        const int ka = k + half * 2;
        v2f a, b;
        a.x = xrow[ka];
        a.y = xrow[ka + 1];
        b.x = W[(size_t)ka * HCW + n];
        b.y = W[(size_t)(ka + 1) * HCW + n];
        acc = __builtin_amdgcn_wmma_f32_16x16x4_f32(
                  false, a, false, b, (short)0, acc, false, false);
    }

    const int mrow = m0 + half * 8;
    float* orow = Hout + (size_t)mrow * HCW + n;
    #pragma unroll
    for (int r = 0; r < 8; ++r)
        if (mrow + r < N) orow[(size_t)r * HCW] = acc[r];
}

// ---------------------------------------------------------------------------
// alpha_s[n,h] = dot(h[n,h,:], att_src[h,:]); same for alpha_d.
// One thread per (node, head).
// ---------------------------------------------------------------------------
__global__ void alpha_kernel(const float* __restrict__ H,
                             const float* __restrict__ a_src,
                             const float* __restrict__ a_dst,
                             float* __restrict__ as, float* __restrict__ ad, int N) {
    int t = blockIdx.x * blockDim.x + threadIdx.x;
    if (t >= N * HEADS) return;
    int n = t >> 2, h = t & 3;
    const float* hp = H + (size_t)n * HCW + h * CH;
    const float* s = a_src + h * CH;
    const float* d = a_dst + h * CH;
    float ss = 0.f, dd = 0.f;
    #pragma unroll 8
    for (int c = 0; c < CH; ++c) { float v = hp[c]; ss += v * s[c]; dd += v * d[c]; }
    as[t] = ss; ad[t] = dd;
}

__global__ void init_kernel(float* __restrict__ agg, float* __restrict__ m,
                            float* __restrict__ denom, int N) {
    int i = blockIdx.x * blockDim.x + threadIdx.x;
    if (i < N * HCW) agg[i] = 0.f;
    if (i < N * HEADS) { m[i] = -__builtin_inff(); denom[i] = 0.f; }
}

__device__ __forceinline__ void edge_sd(const long long* __restrict__ ei,
                                        int E, int idx, int& s, int& d) {
    if (idx < E) { s = (int)ei[idx]; d = (int)ei[E + idx]; }
    else { s = idx - E; d = idx - E; }          // self-loops appended
}

__device__ __forceinline__ void atomicMaxF32(float* addr, float v) {
    if (v >= 0.f) atomicMax((int*)addr, __float_as_int(v));
    else          atomicMin((unsigned int*)addr, __float_as_uint(v));
}

// e = leaky_relu(alpha_s[src] + alpha_d[dst]); running segment max into m[dst].
__global__ void edge_max_kernel(const long long* __restrict__ ei, int E, int ET,
                                const float* __restrict__ as, const float* __restrict__ ad,
                                float* __restrict__ ebuf, float* __restrict__ m) {
    int e = blockIdx.x * blockDim.x + threadIdx.x;
    if (e >= ET) return;
    int s, d; edge_sd(ei, E, e, s, d);
    #pragma unroll
    for (int h = 0; h < HEADS; ++h) {
        float v = as[s * HEADS + h] + ad[d * HEADS + h];
        v = (v >= 0.f) ? v : LRELU_SLOPE * v;
        ebuf[(size_t)e * HEADS + h] = v;
        atomicMaxF32(&m[d * HEADS + h], v);
    }
}

// w = exp(e - m[dst]); denom[dst] += w.   (w overwrites ebuf in place)
__global__ void edge_exp_kernel(const long long* __restrict__ ei, int E, int ET,
                                float* __restrict__ ebuf,
                                const float* __restrict__ m, float* __restrict__ denom) {
    int e = blockIdx.x * blockDim.x + threadIdx.x;
    if (e >= ET) return;
    int s, d; edge_sd(ei, E, e, s, d);
    (void)s;
    #pragma unroll
    for (int h = 0; h < HEADS; ++h) {
        float w = __expf(ebuf[(size_t)e * HEADS + h] - m[d * HEADS + h]);
        ebuf[(size_t)e * HEADS + h] = w;
        atomicAdd(&denom[d * HEADS + h], w);
    }
}

// agg[dst, c] += (w / denom[dst]) * h[src, c] — thread handles 4 channels.
// h, agg, denom are all L2-resident (51 MB + 51 MB + <1 MB << 192 MB L2).
__global__ void aggregate_kernel(const long long* __restrict__ ei, int E, int ET,
                                 const float* __restrict__ H,
                                 const float* __restrict__ w,
                                 const float* __restrict__ denom,
                                 float* __restrict__ agg) {
    long long t = (long long)blockIdx.x * blockDim.x + threadIdx.x;
    if (t >= (long long)ET * 64) return;
    int e  = (int)(t >> 6);
    int c4 = ((int)t & 63) * 4;           // channel base, 0..252
    int s, d; edge_sd(ei, E, e, s, d);
    int h = c4 >> 6;                      // head index
    float coef = w[(size_t)e * HEADS + h] / (denom[d * HEADS + h] + 1e-16f);
    const float4 hv = *(const float4*)(H + (size_t)s * HCW + c4);
    float* ap = agg + (size_t)d * HCW + c4;
    atomicAdd(ap + 0, coef * hv.x);
    atomicAdd(ap + 1, coef * hv.y);
    atomicAdd(ap + 2, coef * hv.z);
    atomicAdd(ap + 3, coef * hv.w);
}

// BN(eval) + ELU; writes next layer's input.
__global__ void epilogue_kernel(const float* __restrict__ agg, const float* __restrict__ bias,
                                const float* __restrict__ g, const float* __restrict__ b,
                                const float* __restrict__ rm, const float* __restrict__ rv,
                                float* __restrict__ xout, int N) {
    int i = blockIdx.x * blockDim.x + threadIdx.x;
    if (i >= N * HCW) return;
    int c = i & (HCW - 1);
    float v = agg[i] + bias[c];
    v = (v - rm[c]) * rsqrtf(rv[c] + BN_EPS) * g[c] + b[c];
    v = (v > 0.f) ? v : (__expf(v) - 1.f);      // ELU
    xout[i] = v;
}

// out[n] = sigmoid(dot(h[n,:], W_out) + b_out). One wave32 per node.
__global__ __launch_bounds__(256)
void out_kernel(const float* __restrict__ X, const float* __restrict__ Wout,
                const float* __restrict__ bout, float* __restrict__ out, int N) {
    int wave = (int)((blockIdx.x * blockDim.x + threadIdx.x) >> 5);
    int lane = threadIdx.x & 31;
    if (wave >= N) return;
    const float* xr = X + (size_t)wave * HCW;
    float sum = 0.f;
    #pragma unroll
    for (int c = lane; c < HCW; c += 32) sum += xr[c] * Wout[c];
    #pragma unroll
    for (int off = 16; off; off >>= 1) sum += __shfl_xor(sum, off, 32);
    if (lane == 0) out[wave] = 1.f / (1.f + __expf(-(sum + bout[0])));
}

// ---------------------------------------------------------------------------
// Input order (dict-insertion): x, edge_index, then per layer l:
//   W{l}, att_src{l}, att_dst{l}, bias{l}, bn_g{l}, bn_b{l}, bn_rm{l}, bn_rv{l}
// then W_out, b_out.
// ---------------------------------------------------------------------------
extern "C" void kernel_launch(void* const* d_in, const int* in_sizes, int n_in,
                              void* d_out, int out_size, void* d_ws, size_t ws_size,
                              hipStream_t stream) {
    const float* x0 = (const float*)d_in[0];
    const long long* ei = (const long long*)d_in[1];
    const int N  = in_sizes[0] / 128;     // 50000
    const int E  = in_sizes[1] / 2;       // 800000
    const int ET = E + N;                 // + self loops

    float* ws     = (float*)d_ws;
    float* buf_x  = ws;                               // [N, 256]
    float* buf_h  = buf_x + (size_t)N * HCW;          // [N, 256]
    float* agg    = buf_h + (size_t)N * HCW;          // [N, 256]
    float* as     = agg   + (size_t)N * HCW;          // [N, 4]
    float* ad     = as    + (size_t)N * HEADS;        // [N, 4]
    float* mb     = ad    + (size_t)N * HEADS;        // [N, 4]
    float* db     = mb    + (size_t)N * HEADS;        // [N, 4]
    float* eb     = db    + (size_t)N * HEADS;        // [ET, 4]

    const float* cur = x0;
    int K = 128;
    for (int l = 0; l < 3; ++l) {
        const float* W    = (const float*)d_in[2 + l * 8 + 0];
        const float* asrc = (const float*)d_in[2 + l * 8 + 1];
        const float* adst = (const float*)d_in[2 + l * 8 + 2];
        const float* bias = (const float*)d_in[2 + l * 8 + 3];
        const float* bng  = (const float*)d_in[2 + l * 8 + 4];
        const float* bnb  = (const float*)d_in[2 + l * 8 + 5];
        const float* bnrm = (const float*)d_in[2 + l * 8 + 6];
        const float* bnrv = (const float*)d_in[2 + l * 8 + 7];

        dim3 ggrid((N + 15) / 16, HCW / 16);
        gemm_wmma_f32<<<ggrid, 32, 0, stream>>>(cur, W, buf_h, N, K);
        alpha_kernel<<<(N * HEADS + 255) / 256, 256, 0, stream>>>(buf_h, asrc, adst, as, ad, N);
        init_kernel<<<(N * HCW + 255) / 256, 256, 0, stream>>>(agg, mb, db, N);
        edge_max_kernel<<<(ET + 255) / 256, 256, 0, stream>>>(ei, E, ET, as, ad, eb, mb);
        edge_exp_kernel<<<(ET + 255) / 256, 256, 0, stream>>>(ei, E, ET, eb, mb, db);
        long long aggT = (long long)ET * 64;
        aggregate_kernel<<<(unsigned)((aggT + 255) / 256), 256, 0, stream>>>(
            ei, E, ET, buf_h, eb, db, agg);
        epilogue_kernel<<<(N * HCW + 255) / 256, 256, 0, stream>>>(
            agg, bias, bng, bnb, bnrm, bnrv, buf_x, N);
        cur = buf_x;
        K = HCW;
    }

    const float* Wout = (const float*)d_in[26];
    const float* bout = (const float*)d_in[27];
    out_kernel<<<(N * 32 + 255) / 256, 256, 0, stream>>>(cur, Wout, bout, (float*)d_out, N);
}